// SoftPerspectiveShader_10007273800423
// MI455X (gfx1250) — compile-verified
//
#include <hip/hip_runtime.h>

// ---- constants from the reference ----
#define KFRAG 8
__device__ __forceinline__ float frcp(float x) { return __builtin_amdgcn_rcpf(x); }

typedef float __attribute__((ext_vector_type(4))) f32x4;
typedef int   __attribute__((ext_vector_type(4))) i32x4;

__global__ __launch_bounds__(256)
void soft_rgb_blend_kernel(const int*   __restrict__ pix_to_face,
                           const float* __restrict__ bary,
                           const float* __restrict__ zbuf,
                           const float* __restrict__ dists,
                           const float* __restrict__ face_colors,
                           float*       __restrict__ out,
                           int n_pix)
{
    const float SIGMA_INV = 1.0e4f;            // 1/1e-4
    const float GAMMA_INV = 1.0e4f;            // 1/1e-4
    const float ZFAR      = 100.0f;
    const float INV_RANGE = 1.0f / 99.0f;      // 1/(zfar-znear)
    const float EPS       = 1.0e-10f;

    int p = blockIdx.x * blockDim.x + threadIdx.x;
    if (p >= n_pix) return;

    // ---- streamed, coalesced 128-bit NT loads; 32-bit indexing so the
    //      backend emits saddr + voffset global_load_b128 th:TH_LOAD_NT ----
    const i32x4* pf4 = (const i32x4*)pix_to_face;   // 2 vec4 per pixel
    i32x4 pfa = __builtin_nontemporal_load(pf4 + 2 * p);
    i32x4 pfb = __builtin_nontemporal_load(pf4 + 2 * p + 1);
    int pf[KFRAG] = {pfa.x, pfa.y, pfa.z, pfa.w, pfb.x, pfb.y, pfb.z, pfb.w};

    // 32-bit gather offsets: idx < 200000 -> off < 1.8M, fits easily.
    // v_max_i32 + v_mul_lo_u32, no 64-bit mul, no divergent branches.
    unsigned off[KFRAG];
    #pragma unroll
    for (int k = 0; k < KFRAG; ++k) {
        int idx = pf[k] < 0 ? 0 : pf[k];
        off[k] = (unsigned)idx * 9u;
    }

    // gfx1250 prefetch of all gather rows ASAP (table is L2-resident: 7.2MB)
    #pragma unroll
    for (int k = 0; k < KFRAG; ++k)
        __builtin_prefetch(face_colors + off[k], 0, 3);

    const f32x4* z4 = (const f32x4*)zbuf;
    const f32x4* d4 = (const f32x4*)dists;
    f32x4 za = __builtin_nontemporal_load(z4 + 2 * p);
    f32x4 zb = __builtin_nontemporal_load(z4 + 2 * p + 1);
    f32x4 da = __builtin_nontemporal_load(d4 + 2 * p);
    f32x4 db = __builtin_nontemporal_load(d4 + 2 * p + 1);
    float zv[KFRAG] = {za.x, za.y, za.z, za.w, zb.x, zb.y, zb.z, zb.w};
    float dv[KFRAG] = {da.x, da.y, da.z, da.w, db.x, db.y, db.z, db.w};

    const f32x4* b4 = (const f32x4*)bary;           // 6 vec4 per pixel
    float bw[KFRAG * 3];
    #pragma unroll
    for (int i = 0; i < 6; ++i) {
        f32x4 v = __builtin_nontemporal_load(b4 + 6 * p + i);
        bw[i * 4 + 0] = v.x; bw[i * 4 + 1] = v.y;
        bw[i * 4 + 2] = v.z; bw[i * 4 + 3] = v.w;
    }

    // ---- pass 1: per-fragment prob, z_inv, barycentric texel; track z_inv_max ----
    float prob[KFRAG], zin[KFRAG], tr[KFRAG], tg[KFRAG], tb[KFRAG];
    float zmax = EPS;     // max(max_k z_inv, EPS)
    float keep = 1.0f;    // prod(1 - prob) == 1 - alpha

    #pragma unroll
    for (int k = 0; k < KFRAG; ++k) {
        bool valid = pf[k] >= 0;
        const float* row = face_colors + off[k];    // [3 verts][3 ch]

        float w0 = bw[k * 3 + 0], w1 = bw[k * 3 + 1], w2 = bw[k * 3 + 2];
        // texel_c = sum_v bary[v] * colors[v][c]
        tr[k] = w0 * row[0] + w1 * row[3] + w2 * row[6];
        tg[k] = w0 * row[1] + w1 * row[4] + w2 * row[7];
        tb[k] = w0 * row[2] + w1 * row[5] + w2 * row[8];

        // prob = sigmoid(-d/sigma) * mask = mask / (1 + exp(d/sigma))
        float pr = valid ? frcp(1.0f + __expf(dv[k] * SIGMA_INV)) : 0.0f;
        prob[k] = pr;
        keep *= (1.0f - pr);

        float zi = valid ? (ZFAR - zv[k]) * INV_RANGE : 0.0f;
        zin[k] = zi;
        zmax = fmaxf(zmax, zi);
    }

    // ---- pass 2: softmax weights (args <= 0, no overflow) and blend ----
    float wsum = 0.0f, cr = 0.0f, cg = 0.0f, cb = 0.0f;
    #pragma unroll
    for (int k = 0; k < KFRAG; ++k) {
        float w = prob[k] * __expf((zin[k] - zmax) * GAMMA_INV);
        wsum += w;
        cr += w * tr[k];
        cg += w * tg[k];
        cb += w * tb[k];
    }

    float delta  = fmaxf(__expf((EPS - zmax) * GAMMA_INV), EPS);
    float invden = frcp(wsum + delta);

    f32x4 o;
    o.x = (cr + delta) * invden;   // background = (1,1,1)
    o.y = (cg + delta) * invden;
    o.z = (cb + delta) * invden;
    o.w = keep;                    // 1 - alpha

    __builtin_nontemporal_store(o, (f32x4*)out + p);
}

extern "C" void kernel_launch(void* const* d_in, const int* in_sizes, int n_in,
                              void* d_out, int out_size, void* d_ws, size_t ws_size,
                              hipStream_t stream) {
    const int*   pix_to_face = (const int*)  d_in[0];
    const float* bary        = (const float*)d_in[1];
    const float* zbuf        = (const float*)d_in[2];
    const float* dists       = (const float*)d_in[3];
    const float* face_colors = (const float*)d_in[4];
    float*       out         = (float*)d_out;

    int n_pix = in_sizes[0] / KFRAG;   // N*H*W
    int block = 256;                   // 8 wave32 waves per block
    int grid  = (n_pix + block - 1) / block;

    soft_rgb_blend_kernel<<<grid, block, 0, stream>>>(
        pix_to_face, bary, zbuf, dists, face_colors, out, n_pix);
}